// RingWiseAgg_60026462929396
// MI455X (gfx1250) — compile-verified
//
#include <hip/hip_runtime.h>
#include <math.h>

typedef __attribute__((ext_vector_type(2))) float v2f;
typedef __attribute__((ext_vector_type(8))) float v8f;

#define DD   128
#define RP1  5

#if defined(__gfx1250__) && __has_builtin(__builtin_amdgcn_wmma_f32_16x16x4_f32)
#define HAVE_WMMA_F32 1
#else
#define HAVE_WMMA_F32 0
#endif

__device__ __forceinline__ v8f wmma_f32_16x16x4(v2f a, v2f b, v8f c) {
#if HAVE_WMMA_F32
    return __builtin_amdgcn_wmma_f32_16x16x4_f32(false, a, false, b, (short)0, c,
                                                 false, false);
#else
    // Host-pass / missing-builtin placeholder (never executed on gfx1250 if builtin exists)
    return c;
#endif
}

// ---------------------------------------------------------------------------
// C = A @ W^T   (A: [M,128] row-major, W: [128,128] row-major)
// Optional per-row input scale  sA = scaleA[row / scaleA_div]   (in_deg for z)
// Optional per-row output scale sO = scaleOut[row]              (out_deg for h)
// Each wave: 16x64 strip via 4 x V_WMMA_F32_16X16X4_F32 accumulators.
// Block = 256 threads = 8 waves -> 64 rows x 128 cols.
// ---------------------------------------------------------------------------
__global__ __launch_bounds__(256)
void wmma_gemm_xwt(const float* __restrict__ A, const float* __restrict__ W,
                   float* __restrict__ out, int mtiles, long out_stride,
                   const float* __restrict__ scaleA, int scaleA_div,
                   const float* __restrict__ scaleOut)
{
    const int lane  = threadIdx.x & 31;
    const int wave  = threadIdx.x >> 5;
    const int mt    = blockIdx.x * 4 + (wave >> 1);
    const int nhalf = wave & 1;
    if (mt >= mtiles) return;

    const int lmod = lane & 15;
    const int lhi  = lane >> 4;                 // 0 or 1 -> K sub-pair
    const int arow = mt * 16 + lmod;
    const float sA = scaleA ? scaleA[arow / scaleA_div] : 1.0f;

    const float2* A2 = (const float2*)(A + (size_t)arow * DD);
    const int ncol   = nhalf * 64 + lmod;
    const float2* B0 = (const float2*)(W + (size_t)(ncol +  0) * DD);
    const float2* B1 = (const float2*)(W + (size_t)(ncol + 16) * DD);
    const float2* B2 = (const float2*)(W + (size_t)(ncol + 32) * DD);
    const float2* B3 = (const float2*)(W + (size_t)(ncol + 48) * DD);

    v8f acc0 = {}, acc1 = {}, acc2 = {}, acc3 = {};

#pragma unroll 8
    for (int k = 0; k < DD; k += 4) {
        const int p = (k >> 1) + lhi;           // float2 index: k + lhi*2
        float2 af = A2[p];
        v2f a; a.x = af.x * sA; a.y = af.y * sA;
        float2 b0f = B0[p], b1f = B1[p], b2f = B2[p], b3f = B3[p];
        v2f b0; b0.x = b0f.x; b0.y = b0f.y;
        v2f b1; b1.x = b1f.x; b1.y = b1f.y;
        v2f b2; b2.x = b2f.x; b2.y = b2f.y;
        v2f b3; b3.x = b3f.x; b3.y = b3f.y;
        acc0 = wmma_f32_16x16x4(a, b0, acc0);
        acc1 = wmma_f32_16x16x4(a, b1, acc1);
        acc2 = wmma_f32_16x16x4(a, b2, acc2);
        acc3 = wmma_f32_16x16x4(a, b3, acc3);
    }

    const int colbase = nhalf * 64 + lmod;
#pragma unroll
    for (int v = 0; v < 8; ++v) {
        const int row = mt * 16 + v + 8 * lhi;  // C/D layout: VGPR v, lane half
        const float sO = scaleOut ? scaleOut[row] : 1.0f;
        float* o = out + (size_t)row * out_stride + colbase;
        o[ 0] = acc0[v] * sO;
        o[16] = acc1[v] * sO;
        o[32] = acc2[v] * sO;
        o[48] = acc3[v] * sO;
    }
}

// ---------------------------------------------------------------------------
__global__ __launch_bounds__(256)
void zero_f32(float* __restrict__ p, long n)
{
    long i = (long)blockIdx.x * blockDim.x + threadIdx.x;
    const long stride = (long)gridDim.x * blockDim.x;
    for (; i < n; i += stride) p[i] = 0.0f;
}

// ---------------------------------------------------------------------------
// z[row, (ring+1)*128 + j] += h[ring, col, j]  for every edge.  One wave/edge,
// 4 floats per lane (float4 load + 4 L2 float atomics, no return -> STOREcnt).
// ---------------------------------------------------------------------------
__global__ __launch_bounds__(256)
void scatter_edges(const int* __restrict__ rows, const int* __restrict__ cols,
                   const float* __restrict__ h, float* __restrict__ z,
                   int nnz_total, int nnz_per_ring, int N)
{
    const int wid  = (int)((blockIdx.x * (long)blockDim.x + threadIdx.x) >> 5);
    const int lane = threadIdx.x & 31;
    if (wid >= nnz_total) return;
    const int ring = wid / nnz_per_ring;
    const int e    = wid - ring * nnz_per_ring;
    const long ei  = (long)ring * nnz_per_ring + e;
    const int row  = rows[ei];
    const int col  = cols[ei];

    const float4 v = *(const float4*)(h + ((size_t)ring * N + col) * DD + lane * 4);
    float* dst = z + (size_t)row * (RP1 * DD) + (ring + 1) * DD + lane * 4;
    atomicAdd(dst + 0, v.x);
    atomicAdd(dst + 1, v.y);
    atomicAdd(dst + 2, v.z);
    atomicAdd(dst + 3, v.w);
}

// ---------------------------------------------------------------------------
// Per-node heterogeneous-sensitivity attention.  One wave32 per node; each
// lane owns 4 contiguous D-elements.  5x5 Gram matrices reduced via shfl_xor.
// ---------------------------------------------------------------------------
__global__ __launch_bounds__(256)
void hetero_attn(const float* __restrict__ ZC, const float* __restrict__ ZD,
                 const float* __restrict__ att_w, const float* __restrict__ att_b,
                 float* __restrict__ out, int N)
{
    const int lane = threadIdx.x & 31;
    const int node = (int)((blockIdx.x * (long)blockDim.x + threadIdx.x) >> 5);
    if (node >= N) return;
    const int jb = lane * 4;

    float zc[RP1][4], zd[RP1][4];
#pragma unroll
    for (int r = 0; r < RP1; ++r) {
        const float4 c = *(const float4*)(ZC + ((size_t)node * RP1 + r) * DD + jb);
        zc[r][0] = c.x; zc[r][1] = c.y; zc[r][2] = c.z; zc[r][3] = c.w;
        const float4 d = *(const float4*)(ZD + ((size_t)node * RP1 + r) * DD + jb);
        zd[r][0] = d.x; zd[r][1] = d.y; zd[r][2] = d.z; zd[r][3] = d.w;
    }

    // symmetric Gram partials (15 unique pairs each)
    float pc[15], pg[15];
    int p = 0;
#pragma unroll
    for (int r = 0; r < RP1; ++r)
#pragma unroll
        for (int s = r; s < RP1; ++s) {
            float ac = 0.f, ag = 0.f;
#pragma unroll
            for (int t = 0; t < 4; ++t) {
                ac += zc[r][t] * zc[s][t];
                ag += zd[r][t] * zd[s][t];
            }
            pc[p] = ac; pg[p] = ag; ++p;
        }
#pragma unroll
    for (int off = 16; off > 0; off >>= 1)
#pragma unroll
        for (int q = 0; q < 15; ++q) {
            pc[q] += __shfl_xor(pc[q], off, 32);
            pg[q] += __shfl_xor(pg[q], off, 32);
        }

    float Cm[RP1][RP1], Gm[RP1][RP1];
    p = 0;
#pragma unroll
    for (int r = 0; r < RP1; ++r)
#pragma unroll
        for (int s = r; s < RP1; ++s) {
            Cm[r][s] = pc[p]; Cm[s][r] = pc[p];
            Gm[r][s] = pg[p]; Gm[s][r] = pg[p];
            ++p;
        }

    float ac[RP1][RP1], ad[RP1][RP1];
#pragma unroll
    for (int r = 0; r < RP1; ++r) {
        // softmax over s of com_score[r][s] = Cm[r][s]
        float m = Cm[r][0];
#pragma unroll
        for (int s = 1; s < RP1; ++s) m = fmaxf(m, Cm[r][s]);
        float sum = 0.f;
#pragma unroll
        for (int s = 0; s < RP1; ++s) { ac[r][s] = __expf(Cm[r][s] - m); sum += ac[r][s]; }
        float inv = 1.0f / sum;
#pragma unroll
        for (int s = 0; s < RP1; ++s) ac[r][s] *= inv;

        // dis_score[r][s] = zd[r].(zd[r]-zd[s]) = G[r][r] - G[r][s]
        float dsc[RP1];
        float m2 = -3.0e38f;
#pragma unroll
        for (int s = 0; s < RP1; ++s) { dsc[s] = Gm[r][r] - Gm[r][s]; m2 = fmaxf(m2, dsc[s]); }
        float sum2 = 0.f;
#pragma unroll
        for (int s = 0; s < RP1; ++s) { ad[r][s] = __expf(dsc[s] - m2); sum2 += ad[r][s]; }
        float inv2 = 1.0f / sum2;
#pragma unroll
        for (int s = 0; s < RP1; ++s) ad[r][s] *= inv2;
    }

    float zcom[RP1][4], zdis[RP1][4];
#pragma unroll
    for (int r = 0; r < RP1; ++r)
#pragma unroll
        for (int t = 0; t < 4; ++t) {
            float a = 0.f, b = 0.f;
#pragma unroll
            for (int s = 0; s < RP1; ++s) {
                a += ac[r][s] * zc[s][t];
                b += ad[r][s] * zd[s][t];
            }
            zcom[r][t] = a;
            zdis[r][t] = zd[r][t] - b;   // sum(alpha)=1
        }

    // beta = sigmoid([z_com, z_dis] . att_w + b)
    float bp = 0.f;
#pragma unroll
    for (int r = 0; r < RP1; ++r)
#pragma unroll
        for (int t = 0; t < 4; ++t) {
            bp += zcom[r][t] * att_w[r * DD + jb + t];
            bp += zdis[r][t] * att_w[RP1 * DD + r * DD + jb + t];
        }
#pragma unroll
    for (int off = 16; off > 0; off >>= 1) bp += __shfl_xor(bp, off, 32);
    const float beta = 1.0f / (1.0f + __expf(-(bp + att_b[0])));
    const float omb  = 1.0f - beta;

#pragma unroll
    for (int r = 0; r < RP1; ++r) {
        float4 o;
        o.x = beta * zcom[r][0] + omb * zdis[r][0];
        o.y = beta * zcom[r][1] + omb * zdis[r][1];
        o.z = beta * zcom[r][2] + omb * zdis[r][2];
        o.w = beta * zcom[r][3] + omb * zdis[r][3];
        *(float4*)(out + (size_t)node * (RP1 * DD) + r * DD + jb) = o;
    }
}

// ---------------------------------------------------------------------------
extern "C" void kernel_launch(void* const* d_in, const int* in_sizes, int n_in,
                              void* d_out, int out_size, void* d_ws, size_t ws_size,
                              hipStream_t stream)
{
    const float* x       = (const float*)d_in[0];
    const float* W_self  = (const float*)d_in[1];
    const float* W_rings = (const float*)d_in[2];
    const float* WC      = (const float*)d_in[3];
    const float* WD      = (const float*)d_in[4];
    const float* att_w   = (const float*)d_in[5];
    const float* att_b   = (const float*)d_in[6];
    const float* out_deg = (const float*)d_in[7];
    const float* in_deg  = (const float*)d_in[8];
    const int*   rows    = (const int*)d_in[9];
    const int*   cols    = (const int*)d_in[10];

    const int N   = in_sizes[0] / DD;            // 50000
    const int R   = in_sizes[2] / (DD * DD);     // 4
    const int NNZ = in_sizes[9] / R;             // 800000

    const size_t seg = (size_t)N * RP1 * DD;     // 32M floats per region
    float* zbuf = (float*)d_ws;                  // [N, 5*128]
    float* bufA = zbuf + seg;                    // h (4 rings) -> later ZD [5N,128]
    float* bufB = bufA + seg;                    // ZC [5N,128]

    const int mtilesN  = N / 16;                 // 3125
    const int mtiles5N = (N * RP1) / 16;         // 15625

    // 1) zero z (ring slots must start at 0 before scatter)
    zero_f32<<<2048, 256, 0, stream>>>(zbuf, (long)seg);

    // 2) z_self = x @ W_self^T  -> z[:, 0:128]  (row stride 640)
    wmma_gemm_xwt<<<(mtilesN + 3) / 4, 256, 0, stream>>>(
        x, W_self, zbuf, mtilesN, (long)(RP1 * DD), nullptr, 1, nullptr);

    // 3) h_i = (x @ W_rings[i]^T) * out_deg  -> bufA[i]
    for (int i = 0; i < R; ++i) {
        wmma_gemm_xwt<<<(mtilesN + 3) / 4, 256, 0, stream>>>(
            x, W_rings + (size_t)i * DD * DD, bufA + (size_t)i * N * DD,
            mtilesN, (long)DD, nullptr, 1, out_deg);
    }

    // 4) scatter: z[row, (i+1)*128+j] += h_i[col, j]
    {
        const int nnz_total = R * NNZ;
        const long waves    = nnz_total;
        const long blocks   = (waves * 32 + 255) / 256;
        scatter_edges<<<(unsigned)blocks, 256, 0, stream>>>(
            rows, cols, bufA, zbuf, nnz_total, NNZ, N);
    }

    // 5) ZC = (z * in_deg) @ WC^T   (z viewed as [5N,128], scale per node)
    wmma_gemm_xwt<<<(mtiles5N + 3) / 4, 256, 0, stream>>>(
        zbuf, WC, bufB, mtiles5N, (long)DD, in_deg, RP1, nullptr);

    // 6) ZD = (z * in_deg) @ WD^T   (overwrites h; h is dead now)
    wmma_gemm_xwt<<<(mtiles5N + 3) / 4, 256, 0, stream>>>(
        zbuf, WD, bufA, mtiles5N, (long)DD, in_deg, RP1, nullptr);

    // 7) per-node attention + gate -> d_out [N, 640]
    {
        const long blocks = ((long)N * 32 + 255) / 256;
        hetero_attn<<<(unsigned)blocks, 256, 0, stream>>>(
            bufB, bufA, att_w, att_b, (float*)d_out, N);
    }
}